// xlayer_48095043780967
// MI455X (gfx1250) — compile-verified
//
#include <hip/hip_runtime.h>

typedef __attribute__((ext_vector_type(2))) float v2f;
typedef __attribute__((ext_vector_type(8))) float v8f;

#define BB 8192
#define DD 512
#define OO 256

// ---------------------------------------------------------------------------
// Stage 1: W[i,o] = sum_c kernel[i,c,o] * xval[c]
// Pure bandwidth problem (256 MB of kernel traffic @ 23.3 TB/s ~= 11 us).
// One block per i (512 blocks), threads over o (256) -> every c-iteration the
// block reads 256 consecutive floats (1 KB, perfectly coalesced).
// xval staged once into LDS; 4 accumulators to break the FMA dependency chain.
// ---------------------------------------------------------------------------
__global__ __launch_bounds__(256) void xlayer_fold_xval(
    const float* __restrict__ kern, const float* __restrict__ xval,
    float* __restrict__ W) {
  __shared__ float sxval[DD];
  const int o = threadIdx.x;
  const int i = blockIdx.x;
  for (int c = threadIdx.x; c < DD; c += 256) sxval[c] = xval[c];
  __syncthreads();

  const float* kp = kern + (size_t)i * (DD * OO) + o;
  float a0 = 0.f, a1 = 0.f, a2 = 0.f, a3 = 0.f;
#pragma unroll 4
  for (int c = 0; c < DD; c += 4) {
    a0 = fmaf(kp[(size_t)(c + 0) * OO], sxval[c + 0], a0);
    a1 = fmaf(kp[(size_t)(c + 1) * OO], sxval[c + 1], a1);
    a2 = fmaf(kp[(size_t)(c + 2) * OO], sxval[c + 2], a2);
    a3 = fmaf(kp[(size_t)(c + 3) * OO], sxval[c + 3], a3);
  }
  W[i * OO + o] = (a0 + a1) + (a2 + a3);
}

// ---------------------------------------------------------------------------
// Stage 2: out = x @ W  (8192x512 * 512x256, all f32)
// Full-precision V_WMMA_F32_16X16X4_F32. One wave -> 16x64 output tile
// (4 v8f accumulators, A fragment reused 4x). K loop steps by 4.
//
// 32-bit A 16x4 layout: lane<16: M=lane, vgpr0/1 = K0/K1; lane>=16: K2/K3.
// 32-bit B 4x16 layout: lanes = N, same K striping.
// 32-bit C/D 16x16: vgpr r -> row r + 8*(lane>=16), col = lane&15.
// ---------------------------------------------------------------------------
__global__ __launch_bounds__(256) void xlayer_wmma_gemm(
    const float* __restrict__ x, const float* __restrict__ W,
    float* __restrict__ out) {
  const int lane = threadIdx.x & 31;
  const int wave = (blockIdx.x << 3) + (threadIdx.x >> 5);  // 8 waves/block
  const int NT = OO / 64;                                   // 4 n-tiles
  const int mBase = (wave / NT) * 16;
  const int nBase = (wave % NT) * 64;
  const int half = lane >> 4;   // 0: K0/K1 pair, 1: K2/K3 pair
  const int l16  = lane & 15;

  v8f acc0 = {}, acc1 = {}, acc2 = {}, acc3 = {};

  const float* xrow = x + (size_t)(mBase + l16) * DD + 2 * half;
  const float* wcol = W + nBase + l16;

#pragma unroll 4
  for (int k0 = 0; k0 < DD; k0 += 4) {
    v2f a;
    a.x = xrow[k0];
    a.y = xrow[k0 + 1];

    const int kr = k0 + 2 * half;
    const float* wp0 = wcol + (size_t)kr * OO;  // row kr   of W
    const float* wp1 = wp0 + OO;                // row kr+1 of W
    v2f b0, b1, b2, b3;
    b0.x = wp0[0];  b0.y = wp1[0];
    b1.x = wp0[16]; b1.y = wp1[16];
    b2.x = wp0[32]; b2.y = wp1[32];
    b3.x = wp0[48]; b3.y = wp1[48];

    acc0 = __builtin_amdgcn_wmma_f32_16x16x4_f32(false, a, false, b0,
                                                 (short)0, acc0, false, false);
    acc1 = __builtin_amdgcn_wmma_f32_16x16x4_f32(false, a, false, b1,
                                                 (short)0, acc1, false, false);
    acc2 = __builtin_amdgcn_wmma_f32_16x16x4_f32(false, a, false, b2,
                                                 (short)0, acc2, false, false);
    acc3 = __builtin_amdgcn_wmma_f32_16x16x4_f32(false, a, false, b3,
                                                 (short)0, acc3, false, false);
  }

  float* orow = out + (size_t)(mBase + 8 * half) * OO + nBase + l16;
#pragma unroll
  for (int r = 0; r < 8; ++r) {
    float* p = orow + (size_t)r * OO;
    p[0]  = acc0[r];
    p[16] = acc1[r];
    p[32] = acc2[r];
    p[48] = acc3[r];
  }
}

extern "C" void kernel_launch(void* const* d_in, const int* in_sizes, int n_in,
                              void* d_out, int out_size, void* d_ws, size_t ws_size,
                              hipStream_t stream) {
  const float* x    = (const float*)d_in[0];  // (8192, 512)
  const float* xval = (const float*)d_in[1];  // (512,)
  const float* kern = (const float*)d_in[2];  // (512, 512, 256)
  float* out = (float*)d_out;                 // (8192, 256)
  float* W   = (float*)d_ws;                  // (512, 256) scratch, 512 KB

  // Stage 1: fold xval into kernel -> W (bandwidth-bound, 256 MB stream)
  xlayer_fold_xval<<<DD, 256, 0, stream>>>(kern, xval, W);

  // Stage 2: out = x @ W via f32 WMMA.
  // 512 M-tiles * 4 N-tiles = 2048 waves = 256 blocks of 8 waves.
  const int mTiles = BB / 16;
  const int nTiles = OO / 64;
  const int blocks = (mTiles * nTiles) / 8;
  xlayer_wmma_gemm<<<blocks, 256, 0, stream>>>(x, W, out);
}